// FlangerModule_33457795236493
// MI455X (gfx1250) — compile-verified
//
#include <hip/hip_runtime.h>
#include <math.h>

#define ASG __attribute__((address_space(1)))
#define ASL __attribute__((address_space(3)))

typedef int fl_v4i __attribute__((ext_vector_type(4)));

// ---- problem constants (from reference) ----
#define FL_D        441          // MAX_DELAY
#define FL_DPAD     448          // 14 * 32: pad for branch-free init + mirror slot
#define FL_N        32768        // samples per channel
#define FL_TILE     1024         // samples per double-buffered chunk
#define FL_NCHUNK   (FL_N / FL_TILE)
#define FL_VECIT    (FL_TILE / (32 * 4))   // 8 b128 transfers per lane per array
#define FL_FEEDBACK 0.5f
#define FL_MIX      0.7f
#define FL_DW       441.0f       // D * WIDTH (WIDTH = 1.0)

// ---- CDNA5 async global->LDS path (ASYNCcnt), guarded ----
#if __has_builtin(__builtin_amdgcn_global_load_async_to_lds_b128)
#define FLANGER_ASYNC 1
#else
#define FLANGER_ASYNC 0
#endif

#if FLANGER_ASYNC
#if __has_builtin(__builtin_amdgcn_s_wait_asynccnt)
#define WAIT_ASYNC(n) do { __builtin_amdgcn_s_wait_asynccnt(n); asm volatile("" ::: "memory"); } while (0)
#else
#define WAIT_ASYNC(n) asm volatile("s_wait_asynccnt %0" :: "n"(n) : "memory")
#endif
#else
#define WAIT_ASYNC(n) asm volatile("" ::: "memory")
#endif

__device__ __forceinline__ void load_tile(const float* __restrict__ gx,
                                          const float* __restrict__ gm,
                                          float* lx, float* lm, int lane) {
#if FLANGER_ASYNC
  // 16 async b128 instructions per chunk (8 per array): each moves 32 lanes x 16B.
#pragma unroll
  for (int t = 0; t < FL_VECIT; ++t) {
    __builtin_amdgcn_global_load_async_to_lds_b128(
        (ASG fl_v4i*)(gx + t * 128 + lane * 4),
        (ASL fl_v4i*)(lx + t * 128 + lane * 4), 0, 0);
    __builtin_amdgcn_global_load_async_to_lds_b128(
        (ASG fl_v4i*)(gm + t * 128 + lane * 4),
        (ASL fl_v4i*)(lm + t * 128 + lane * 4), 0, 0);
  }
#else
  // synchronous fallback: coalesced b128 loads + ds stores
#pragma unroll
  for (int t = 0; t < FL_VECIT; ++t) {
    ((float4*)lx)[t * 32 + lane] = ((const float4*)gx)[t * 32 + lane];
    ((float4*)lm)[t * 32 + lane] = ((const float4*)gm)[t * 32 + lane];
  }
#endif
}

// One wave32 workgroup per channel. The 441-tap delay line lives in LDS with
// buf[441] mirroring buf[0], so the fractional-delay gather is two adjacent
// dwords (-> ds_load_2addr_b32). All 32 lanes run the inherently sequential
// scan wave-uniformly: LDS reads broadcast, same-address writes carry
// identical values. Global traffic is kept entirely off the serial chain via
// double-buffered ASYNCcnt global->LDS DMA.
__global__ __launch_bounds__(32) void flanger_kernel(const float* __restrict__ x,
                                                     const float* __restrict__ m,
                                                     float* __restrict__ o) {
  __shared__ float s_buf[FL_DPAD];         // circular delay line (+ mirror slot 441)
  __shared__ float s_x[2][FL_TILE];        // double-buffered dry input
  __shared__ float s_m[2][FL_TILE];        // double-buffered LFO signal
  __shared__ float s_o[FL_TILE];           // staged output tile

  const int lane = threadIdx.x;
  const long long c = blockIdx.x;          // flattened (B, C) channel
  const float* xc = x + c * FL_N;
  const float* mc = m + c * FL_N;
  float*       oc = o + c * FL_N;

  // kick off chunk-0 DMA first so it overlaps the delay-line init
  load_tile(xc, mc, s_x[0], s_m[0], lane);

  // branch-free zero init: 448 = 14 * 32 entries, 14 stores per lane
#pragma unroll
  for (int j = 0; j < FL_DPAD / 32; ++j) s_buf[lane + 32 * j] = 0.0f;
  __syncthreads();

  int widx = 0;                            // global write index, mod D
  for (int k = 0; k < FL_NCHUNK; ++k) {
    const int cur = k & 1;
    if (k + 1 < FL_NCHUNK) {
      // prefetch next chunk, then wait until only its 16 loads are in flight
      // (async loads complete in order -> current chunk is resident)
      load_tile(xc + (k + 1) * FL_TILE, mc + (k + 1) * FL_TILE,
                s_x[cur ^ 1], s_m[cur ^ 1], lane);
      WAIT_ASYNC(16);
    } else {
      WAIT_ASYNC(0);
    }
    __syncthreads();

    const float* lx = s_x[cur];
    const float* lm = s_m[cur];

#pragma unroll 4
    for (int i = 0; i < FL_TILE; ++i) {
      float xv = lx[i];
      float mv = lm[i];
      // read = (widx - D*WIDTH*mod + D) mod D  (value in (0, 2D) -> one cond-sub)
      float rd = (float)widx - FL_DW * mv + 441.0f;
      rd = (rd >= 441.0f) ? (rd - 441.0f) : rd;     // exact (Sterbenz)
      float pf = floorf(rd);
      float fr = rd - pf;
      int pi = (int)pf;                             // 0..440
      float pv = s_buf[pi];                         // adjacent pair: compiler
      float nv = s_buf[pi + 1];                     //   merges to ds_load_2addr
      float interp = fr * nv + (1.0f - fr) * pv;
      float w = xv + FL_FEEDBACK * interp;          // feedback write
      s_buf[widx] = w;
      int wm = (widx == 0) ? FL_D : widx;           // keep mirror buf[441]==buf[0]
      s_buf[wm] = w;                                // (same-value dup store if widx!=0)
      float wet = xv + interp;                      // DEPTH = 1.0
      s_o[i] = (1.0f - FL_MIX) * xv + FL_MIX * wet;
      widx = (widx + 1 >= FL_D) ? 0 : widx + 1;
    }
    __syncthreads();

    // coalesced b128 output store (regular STOREcnt path: keeps ASYNCcnt
    // load-only so the in-order-load wait logic above stays exact)
    float4* go = (float4*)(oc + (size_t)k * FL_TILE);
    const float4* lo = (const float4*)s_o;
#pragma unroll
    for (int t = 0; t < FL_VECIT; ++t)
      go[t * 32 + lane] = lo[t * 32 + lane];
    __syncthreads();
  }
}

extern "C" void kernel_launch(void* const* d_in, const int* in_sizes, int n_in,
                              void* d_out, int out_size, void* d_ws, size_t ws_size,
                              hipStream_t stream) {
  const float* x = (const float*)d_in[0];      // [8, 2, 32768] f32
  const float* m = (const float*)d_in[1];      // [8, 2, 32768] f32
  float* o = (float*)d_out;                    // [8, 2, 32768] f32
  const int channels = in_sizes[0] / FL_N;     // B*C = 16
  flanger_kernel<<<dim3(channels), dim3(32), 0, stream>>>(x, m, o);
}